// GNNEstimator_60524679135241
// MI455X (gfx1250) — compile-verified
//
#include <hip/hip_runtime.h>
#include <hip/hip_bf16.h>
#include <math.h>

typedef __attribute__((ext_vector_type(2))) float v2f;
typedef __attribute__((ext_vector_type(8))) float v8f;

// ---------------------------------------------------------------------------
// Float atomic max via monotonic int/uint bit trick (lowers to
// global_atomic_max_i32 / global_atomic_min_u32 on gfx1250).
// ---------------------------------------------------------------------------
__device__ __forceinline__ void atomicMaxFloat(float* addr, float v) {
  if (v >= 0.0f) {
    atomicMax((int*)addr, __float_as_int(v));
  } else {
    atomicMin((unsigned int*)addr, (unsigned int)__float_as_int(v));
  }
}

// ---------------------------------------------------------------------------
// Generic fill
// ---------------------------------------------------------------------------
__global__ void fill_kernel(float* __restrict__ p, long n, float v) {
  long i = (long)blockIdx.x * blockDim.x + threadIdx.x;
  if (i < n) p[i] = v;
}

// ---------------------------------------------------------------------------
// Out[M, NT*16] = A[M,K] @ W[K, NT*16] + bias  via V_WMMA_F32_16X16X4_F32.
//
// One wave owns one 16-row m-tile and ALL NT n-tiles: NT accumulator
// fragments live in registers, and each A fragment is loaded ONCE per
// K-step then amortized over NT WMMAs (cuts A re-read traffic NT x vs the
// one-tile-per-wave scheme; A is the large operand, W is L1-resident).
//
// A frag : lanes 0-15 -> M=lane, K=k+0/1 ; lanes 16-31 -> M=lane-16, K=k+2/3
// B frag : lanes 0-15 -> N=lane, K=k+0/1 ; lanes 16-31 -> N=lane-16, K=k+2/3
// D frag : VGPR r -> (lanes 0-15: M=r, lanes 16-31: M=r+8), N = lane&15
// m-tile index is CLAMPED (not early-returned) so EXEC stays all-ones
// around every WMMA (ISA requirement).
// ---------------------------------------------------------------------------
template <int K, int NT>
__global__ void gemm_bias_wmma(const float* __restrict__ A,
                               const float* __restrict__ W,
                               const float* __restrict__ bias,
                               float* __restrict__ Out,
                               int Mtiles) {
  constexpr int Nout = NT * 16;
  int lane = threadIdx.x & 31;
  int wave = threadIdx.x >> 5;
  int mt = blockIdx.x * (blockDim.x >> 5) + wave;
  if (mt >= Mtiles) mt = Mtiles - 1;  // clamp: keep EXEC == all 1s

  int r16   = lane & 15;
  int khalf = (lane >> 4) << 1;  // 0 for lanes 0-15, 2 for lanes 16-31

  const float* arow  = A + (long)(mt * 16 + r16) * K + khalf;
  const float* wbase = W + (long)khalf * Nout + r16;

  v8f acc[NT];
#pragma unroll
  for (int nt = 0; nt < NT; ++nt) acc[nt] = (v8f){};

  for (int k = 0; k < K; k += 4) {
    v2f a;
    a.x = arow[k];
    a.y = arow[k + 1];
    const float* wk = wbase + (long)k * Nout;
#pragma unroll
    for (int nt = 0; nt < NT; ++nt) {
      v2f b;
      b.x = wk[nt * 16];
      b.y = wk[nt * 16 + Nout];
      acc[nt] = __builtin_amdgcn_wmma_f32_16x16x4_f32(
          /*neg_a=*/false, a, /*neg_b=*/false, b,
          /*c_mod=*/(short)0, acc[nt], /*reuse_a=*/false, /*reuse_b=*/false);
    }
  }

  int rowBase = mt * 16 + ((lane >> 4) << 3);
#pragma unroll
  for (int nt = 0; nt < NT; ++nt) {
    int col = nt * 16 + r16;
    float bv = bias[col];
#pragma unroll
    for (int r = 0; r < 8; ++r) {
      Out[(long)(rowBase + r) * Nout + col] = acc[nt][r] + bv;
    }
  }
}

// ---------------------------------------------------------------------------
// Per-(edge, head): alpha = att[h] . leaky_relu(XR[dst,h,:] + XL[src,h,:])
// and atomic-max into AMAX[dst,h]. Edges e >= E are self-loops (src=dst=e-E).
// ---------------------------------------------------------------------------
__global__ void edge_score(const int* __restrict__ srcIdx,
                           const int* __restrict__ dstIdx,
                           long E_, long EL,
                           const float* __restrict__ XL,
                           const float* __restrict__ XR,
                           const float* __restrict__ att,
                           float* __restrict__ ALPHA,
                           float* __restrict__ AMAX,
                           int H, int HC) {
  long t = (long)blockIdx.x * blockDim.x + threadIdx.x;
  if (t >= EL * H) return;
  int h = (int)(t % H);
  long e = t / H;
  int s, d;
  if (e < E_) { s = srcIdx[e]; d = dstIdx[e]; }
  else        { s = d = (int)(e - E_); }

  const float4* xl = (const float4*)(XL + (long)s * HC + h * 32);
  const float4* xr = (const float4*)(XR + (long)d * HC + h * 32);
  const float4* at = (const float4*)(att + h * 32);
  float sum = 0.0f;
#pragma unroll
  for (int i = 0; i < 8; ++i) {
    float4 a = xl[i];
    float4 b = xr[i];
    float4 w = at[i];
    float vx = a.x + b.x; vx = vx > 0.0f ? vx : 0.2f * vx;
    float vy = a.y + b.y; vy = vy > 0.0f ? vy : 0.2f * vy;
    float vz = a.z + b.z; vz = vz > 0.0f ? vz : 0.2f * vz;
    float vw = a.w + b.w; vw = vw > 0.0f ? vw : 0.2f * vw;
    sum += vx * w.x + vy * w.y + vz * w.z + vw * w.w;
  }
  ALPHA[t] = sum;  // t == e*H + h
  atomicMaxFloat(&AMAX[(long)d * H + h], sum);
}

// ---------------------------------------------------------------------------
// Per-(edge, head): alpha = exp(alpha - amax[dst]); atomic-add into denom.
// ---------------------------------------------------------------------------
__global__ void edge_exp(const int* __restrict__ dstIdx,
                         long E_, long EL,
                         float* __restrict__ ALPHA,
                         const float* __restrict__ AMAX,
                         float* __restrict__ DENOM,
                         int H) {
  long t = (long)blockIdx.x * blockDim.x + threadIdx.x;
  if (t >= EL * H) return;
  int h = (int)(t % H);
  long e = t / H;
  int d = (e < E_) ? dstIdx[e] : (int)(e - E_);
  float a = expf(ALPHA[t] - AMAX[(long)d * H + h]);
  ALPHA[t] = a;
  atomicAdd(&DENOM[(long)d * H + h], a);
}

// ---------------------------------------------------------------------------
// Per-(edge, head): OUT[dst,h,:] += (alpha/denom) * XL[src,h,:]  (scatter-add)
// ---------------------------------------------------------------------------
__global__ void edge_aggregate(const int* __restrict__ srcIdx,
                               const int* __restrict__ dstIdx,
                               long E_, long EL,
                               const float* __restrict__ ALPHA,
                               const float* __restrict__ DENOM,
                               const float* __restrict__ XL,
                               float* __restrict__ OUT,
                               int H, int HC) {
  long t = (long)blockIdx.x * blockDim.x + threadIdx.x;
  if (t >= EL * H) return;
  int h = (int)(t % H);
  long e = t / H;
  int s, d;
  if (e < E_) { s = srcIdx[e]; d = dstIdx[e]; }
  else        { s = d = (int)(e - E_); }
  float a = ALPHA[t] / (DENOM[(long)d * H + h] + 1e-16f);
  const float* xl = XL + (long)s * HC + h * 32;
  float* o = OUT + (long)d * HC + h * 32;
#pragma unroll
  for (int c = 0; c < 32; ++c) {
    atomicAdd(&o[c], a * xl[c]);
  }
}

// ---------------------------------------------------------------------------
// H[i] = relu(H[i] + bias[i % HC])
// ---------------------------------------------------------------------------
__global__ void bias_relu(float* __restrict__ Hbuf, const float* __restrict__ bias,
                          long n, int HC) {
  long i = (long)blockIdx.x * blockDim.x + threadIdx.x;
  if (i < n) {
    float v = Hbuf[i] + bias[(int)(i % HC)];
    Hbuf[i] = v > 0.0f ? v : 0.0f;
  }
}

// ---------------------------------------------------------------------------
// MLP head per node: 32 -> relu(16) -> 2; sigmoid/tanh output transforms.
// ---------------------------------------------------------------------------
__global__ void head_kernel(const float* __restrict__ H3,
                            const float* __restrict__ wm1, const float* __restrict__ bm1,
                            const float* __restrict__ wm2, const float* __restrict__ bm2,
                            float* __restrict__ out, int N) {
  int n = blockIdx.x * blockDim.x + threadIdx.x;
  if (n >= N) return;
  const float* h = H3 + (long)n * 32;
  float m[16];
#pragma unroll
  for (int j = 0; j < 16; ++j) m[j] = bm1[j];
  for (int i = 0; i < 32; ++i) {
    float hi = h[i];
#pragma unroll
    for (int j = 0; j < 16; ++j) m[j] += hi * wm1[i * 16 + j];
  }
  float r0 = bm2[0], r1 = bm2[1];
#pragma unroll
  for (int j = 0; j < 16; ++j) {
    float mj = m[j] > 0.0f ? m[j] : 0.0f;
    r0 += mj * wm2[j * 2 + 0];
    r1 += mj * wm2[j * 2 + 1];
  }
  out[2 * n + 0] = 1.0f / (1.0f + expf(-r0)) + 0.5f;
  out[2 * n + 1] = tanhf(r1) * 180.0f;
}

// ---------------------------------------------------------------------------
// Host orchestration
// ---------------------------------------------------------------------------
extern "C" void kernel_launch(void* const* d_in, const int* in_sizes, int n_in,
                              void* d_out, int out_size, void* d_ws, size_t ws_size,
                              hipStream_t stream) {
  const float* x   = (const float*)d_in[0];
  const int*   ei  = (const int*)d_in[1];
  const float* w1l = (const float*)d_in[2];
  const float* b1l = (const float*)d_in[3];
  const float* w1r = (const float*)d_in[4];
  const float* b1r = (const float*)d_in[5];
  const float* a1  = (const float*)d_in[6];
  const float* c1  = (const float*)d_in[7];
  const float* w2l = (const float*)d_in[8];
  const float* b2l = (const float*)d_in[9];
  const float* w2r = (const float*)d_in[10];
  const float* b2r = (const float*)d_in[11];
  const float* a2  = (const float*)d_in[12];
  const float* c2  = (const float*)d_in[13];
  const float* w3l = (const float*)d_in[14];
  const float* b3l = (const float*)d_in[15];
  const float* w3r = (const float*)d_in[16];
  const float* b3r = (const float*)d_in[17];
  const float* a3  = (const float*)d_in[18];
  const float* c3  = (const float*)d_in[19];
  const float* wm1 = (const float*)d_in[20];
  const float* bm1 = (const float*)d_in[21];
  const float* wm2 = (const float*)d_in[22];
  const float* bm2 = (const float*)d_in[23];

  const long Nn = in_sizes[0] / 64;   // 100000
  const long E_ = in_sizes[1] / 2;    // 1600000
  const long EL = E_ + Nn;            // edges + self-loops
  const int* srcIdx = ei;             // edge_index row 0
  const int* dstIdx = ei + E_;        // edge_index row 1

  // Workspace carving (256B aligned)
  char* ws = (char*)d_ws;
  auto carve = [&](size_t bytes) -> float* {
    float* p = (float*)ws;
    ws += (bytes + 255) & ~(size_t)255;
    return p;
  };
  float* XL    = carve((size_t)Nn * 128 * sizeof(float));
  float* XR    = carve((size_t)Nn * 128 * sizeof(float));
  float* HA    = carve((size_t)Nn * 128 * sizeof(float));
  float* HB    = carve((size_t)Nn * 64 * sizeof(float));
  float* ALPHA = carve((size_t)EL * 4 * sizeof(float));
  float* AMAX  = carve((size_t)Nn * 4 * sizeof(float));
  float* DENOM = carve((size_t)Nn * 4 * sizeof(float));
  (void)ws_size; (void)n_in; (void)out_size;

  const int Mtiles = (int)((Nn + 15) / 16);        // 6250
  const int gemmBlocks = (Mtiles + 7) / 8;         // 8 waves / 256-thread block

  // Shared attention phase for all three layers.
  auto runAttention = [&](int H, const float* att, const float* bias, float* OUT) {
    const int HC = H * 32;
    const long nh  = Nn * H;
    const long nhc = Nn * HC;
    fill_kernel<<<(int)((nh + 255) / 256), 256, 0, stream>>>(AMAX, nh, -INFINITY);
    fill_kernel<<<(int)((nh + 255) / 256), 256, 0, stream>>>(DENOM, nh, 0.0f);
    fill_kernel<<<(int)((nhc + 255) / 256), 256, 0, stream>>>(OUT, nhc, 0.0f);

    const long EH = EL * H;
    const int eb = (int)((EH + 255) / 256);
    edge_score<<<eb, 256, 0, stream>>>(srcIdx, dstIdx, E_, EL, XL, XR, att, ALPHA, AMAX, H, HC);
    edge_exp<<<eb, 256, 0, stream>>>(dstIdx, E_, EL, ALPHA, AMAX, DENOM, H);
    edge_aggregate<<<eb, 256, 0, stream>>>(srcIdx, dstIdx, E_, EL, ALPHA, DENOM, XL, OUT, H, HC);
    bias_relu<<<(int)((nhc + 255) / 256), 256, 0, stream>>>(OUT, bias, nhc, HC);
  };

  // conv1: 64 -> 4 heads x 32 (concat -> 128)
  gemm_bias_wmma<64, 8><<<gemmBlocks, 256, 0, stream>>>(x, w1l, b1l, XL, Mtiles);
  gemm_bias_wmma<64, 8><<<gemmBlocks, 256, 0, stream>>>(x, w1r, b1r, XR, Mtiles);
  runAttention(4, a1, c1, HA);

  // conv2: 128 -> 2 heads x 32 (concat -> 64)
  gemm_bias_wmma<128, 4><<<gemmBlocks, 256, 0, stream>>>(HA, w2l, b2l, XL, Mtiles);
  gemm_bias_wmma<128, 4><<<gemmBlocks, 256, 0, stream>>>(HA, w2r, b2r, XR, Mtiles);
  runAttention(2, a2, c2, HB);

  // conv3: 64 -> 1 head x 32 (mean over 1 head == identity)
  gemm_bias_wmma<64, 2><<<gemmBlocks, 256, 0, stream>>>(HB, w3l, b3l, XL, Mtiles);
  gemm_bias_wmma<64, 2><<<gemmBlocks, 256, 0, stream>>>(HB, w3r, b3r, XR, Mtiles);
  runAttention(1, a3, c3, HA);

  head_kernel<<<(int)((Nn + 255) / 256), 256, 0, stream>>>(
      HA, wm1, bm1, wm2, bm2, (float*)d_out, (int)Nn);
}